// Pytorch3dRasterizer_5669356832405
// MI455X (gfx1250) — compile-verified
//
#include <hip/hip_runtime.h>
#include <stdint.h>

// MI455X (gfx1250) fragment-attribute interpolation (pytorch3d rasterizer).
// HBM-bound random gather: ~1.1 GB total -> ~47us floor at 23.3 TB/s.
// Design: 2 tiles/wave software pipeline on ASYNCcnt partial waits
// (global_load_async_to_lds_b128), WMMA f32 16x16x4 contraction with a
// compile-time block-diagonal barycentric selector, coalesced NCHW stores.

typedef __attribute__((ext_vector_type(2))) float v2f;
typedef __attribute__((ext_vector_type(8))) float v8f;

namespace {
constexpr int kH = 512, kW = 512, kD = 64;
constexpr int kRowF  = 72;                 // LDS floats per K-row: 288B, 16B-aligned, bank-padded
constexpr int kTileF = 48 * kRowF;         // 48 K-rows (16 px x 3 verts) per tile
constexpr int kWavesPerBlock = 2;
constexpr int kTiles = 4 * kH * (kW / 16); // 65536 tiles of 16 pixels along W
}

__global__ __launch_bounds__(64)
void frag_interp_kernel(const float* __restrict__ attr,
                        const long long* __restrict__ p2f,
                        const float* __restrict__ bary,
                        float* __restrict__ out) {
  __shared__ float sh[kWavesPerBlock * 2 * kTileF];  // 55296 B

  const int lane = threadIdx.x & 31;
  const int wid  = threadIdx.x >> 5;
  const int p    = lane & 15;
  const bool hi  = lane >= 16;

  const int wg    = blockIdx.x * kWavesPerBlock + wid;  // [0, 32768)
  const int tile0 = wg * 2;                             // even -> tile1 shares (n, h)
  const int n  = tile0 >> 14;
  const int h  = (tile0 >> 5) & (kH - 1);
  const int w0 = (tile0 & 31) << 4;                     // tile1 at w0 + 16

  // ---- header: lane owns pixel (n, h, w0 + lane); lanes 0-15 = tile0, 16-31 = tile1 ----
  const size_t pixIdx = ((size_t)n * kH + h) * kW + (w0 + lane);
  const long long face = p2f[pixIdx];
  const float visf = (face >= 0) ? 1.0f : 0.0f;
  const int fidx = (face < 0) ? 0 : (int)face;          // clamp background to face 0
  const float b0 = bary[pixIdx * 3 + 0];
  const float b1 = bary[pixIdx * 3 + 1];
  const float b2 = bary[pixIdx * 3 + 2];

  float* shw = sh + wid * (2 * kTileF);
  const uint32_t ldsBase = (uint32_t)(uintptr_t)shw;    // low 32 bits of flat LDS ptr = LDS offset
  const char* attrB = (const char*)attr;

  // ---- async gather: 96 rows x 256B -> LDS; 48 x b128/lane (tile0 first 24) ----
  #pragma unroll
  for (int i = 0; i < 48; ++i) {
    const int id  = i * 32 + lane;       // chunk id in [0, 1536)
    const int row = id >> 4;             // tile*48 + pixel*3 + vert, [0, 96)
    const int cib = id & 15;             // 16B chunk within 256B row
    const int pr  = row / 3;             // == tile*16 + pixel == shfl source lane
    const int cc  = row - pr * 3;
    const int f   = __shfl(fidx, pr, 32);
    const uint64_t ga = (uint64_t)(uintptr_t)
        (attrB + ((size_t)f * 768 + (size_t)cc * 256 + (size_t)cib * 16));
    const uint32_t la = ldsBase + (uint32_t)(row * (kRowF * 4) + cib * 16);
    asm volatile("global_load_async_to_lds_b128 %0, %1, off"
                 :: "v"(la), "v"(ga) : "memory");
  }

  const size_t plane = (size_t)kH * kW;

  auto compute_store = [&](int t) {
    // broadcast tile t's per-pixel bary / visibility to the whole wave
    const int src = p + (t << 4);
    const float c0  = __shfl(b0, src, 32);
    const float c1  = __shfl(b1, src, 32);
    const float c2  = __shfl(b2, src, 32);
    const float vis = __shfl(visf, src, 32);

    const float* shT = shw + t * kTileF;
    const float* rA  = shT + (hi ? 2 * kRowF : 0);      // half-wave K offset for A operand

    v8f a0 = {}, a1 = {}, a2 = {}, a3 = {};
    #pragma unroll
    for (int kc = 0; kc < 12; ++kc) {
      // B (4x16 px): block-diagonal bary selector. K, K/3, K%3 are all
      // compile-time after unroll -> only v_cmp + v_cndmask at runtime.
      float v[4];
      #pragma unroll
      for (int j = 0; j < 4; ++j) {
        const int K  = 4 * kc + j;
        const int PK = K / 3;
        const int C  = K - 3 * PK;
        const float bv = (C == 0) ? c0 : ((C == 1) ? c1 : c2);  // folds
        v[j] = (p == PK) ? bv : 0.0f;
      }
      v2f Bv;
      Bv.x = hi ? v[2] : v[0];
      Bv.y = hi ? v[3] : v[1];

      // A (16 ch x 4): lanes 0-15 hold K=4kc+0/+1 in v0/v1, lanes 16-31 +2/+3
      const float* r0 = rA + 4 * kc * kRowF;
      const float* r1 = r0 + kRowF;
      v2f Av;
      Av.x = r0[p +  0]; Av.y = r1[p +  0];
      a0 = __builtin_amdgcn_wmma_f32_16x16x4_f32(false, Av, false, Bv, (short)0, a0, false, false);
      Av.x = r0[p + 16]; Av.y = r1[p + 16];
      a1 = __builtin_amdgcn_wmma_f32_16x16x4_f32(false, Av, false, Bv, (short)0, a1, false, false);
      Av.x = r0[p + 32]; Av.y = r1[p + 32];
      a2 = __builtin_amdgcn_wmma_f32_16x16x4_f32(false, Av, false, Bv, (short)0, a2, false, false);
      Av.x = r0[p + 48]; Av.y = r1[p + 48];
      a3 = __builtin_amdgcn_wmma_f32_16x16x4_f32(false, Av, false, Bv, (short)0, a3, false, false);
    }

    // D (16 ch x 16 px): VGPR r -> channel (+8 hi half), lane&15 -> pixel.
    // Per VGPR: 16 lanes write 16 consecutive w -> coalesced 64B segments.
    const size_t base = (size_t)n * (kD + 1) * plane + (size_t)h * kW + (w0 + (t << 4)) + p;
    const int dadd = hi ? 8 : 0;
    #pragma unroll
    for (int r = 0; r < 8; ++r) {
      out[base + (size_t)( 0 + r + dadd) * plane] = a0[r] * vis;
      out[base + (size_t)(16 + r + dadd) * plane] = a1[r] * vis;
      out[base + (size_t)(32 + r + dadd) * plane] = a2[r] * vis;
      out[base + (size_t)(48 + r + dadd) * plane] = a3[r] * vis;
    }
  };

  // ---- pipelined: tile0 compute overlaps tile1's in-flight gather ----
  asm volatile("s_wait_asynccnt 0x18" ::: "memory");  // <=24 outstanding: tile0 landed
  compute_store(0);
  asm volatile("s_wait_asynccnt 0x0" ::: "memory");   // tile1 landed
  compute_store(1);

  // ---- visibility channel: 32 lanes, own pixel, one coalesced 128B store ----
  out[((size_t)n * (kD + 1) + kD) * plane + (size_t)h * kW + (w0 + lane)] = visf;
}

extern "C" void kernel_launch(void* const* d_in, const int* in_sizes, int n_in,
                              void* d_out, int out_size, void* d_ws, size_t ws_size,
                              hipStream_t stream) {
  (void)in_sizes; (void)n_in; (void)out_size; (void)d_ws; (void)ws_size;
  const float*     attr = (const float*)d_in[0];      // [4, 80000, 3, 64] f32
  const long long* p2f  = (const long long*)d_in[1];  // [4, 512, 512, 1] i64
  const float*     bary = (const float*)d_in[2];      // [4, 512, 512, 1, 3] f32
  float* out = (float*)d_out;                         // [4, 65, 512, 512] f32

  dim3 grid((kTiles / 2) / kWavesPerBlock);  // 16384
  dim3 block(kWavesPerBlock * 32);           // 64 threads = 2 waves
  hipLaunchKernelGGL(frag_interp_kernel, grid, block, 0, stream, attr, p2f, bary, out);
}